// GCN_60455959658662
// MI455X (gfx1250) — compile-verified
//
#include <hip/hip_runtime.h>

#define NNODES 50000
#define NEDGES 800000
#define HID 100
#define KDIM 100       // contraction dim for every GEMM in this net
#define NTASKS 128
#define NGRAPHS 2000
#define EPSV 1e-16f

typedef float v2f __attribute__((ext_vector_type(2)));
typedef float v4f __attribute__((ext_vector_type(4)));
typedef float v8f __attribute__((ext_vector_type(8)));

__device__ __forceinline__ float clip_inv(float v) {
    // clamp(EPS, 100) then pow(-1)
    return 1.0f / fminf(fmaxf(v, EPSV), 100.0f);
}

// ---------------- zero fill (float4 granularity) ----------------
__global__ void k_zero4(v4f* __restrict__ p, int n4) {
    int i = blockIdx.x * blockDim.x + threadIdx.x;
    if (i < n4) {
        v4f z = {0.f, 0.f, 0.f, 0.f};
        p[i] = z;
    }
}

// ---------------- atom encoder: h[n,c..c+3] = sum_f emb[x[n,f]+off[f], c..c+3] ----------------
__global__ void k_encoder(const int* __restrict__ x, const float* __restrict__ emb,
                          float* __restrict__ h) {
    int idx = blockIdx.x * blockDim.x + threadIdx.x;  // NNODES * 25 work items
    if (idx >= NNODES * (HID / 4)) return;
    int n = idx / (HID / 4);
    int c = (idx - n * (HID / 4)) * 4;
    const int off[9] = {0, 119, 124, 136, 148, 158, 164, 170, 172};
    v4f s = {0.f, 0.f, 0.f, 0.f};
#pragma unroll
    for (int f = 0; f < 9; ++f) {
        int row = x[n * 9 + f] + off[f];
        s += *(const v4f*)(emb + (size_t)row * HID + c);  // 16B aligned: row*400 + c*4
    }
    *(v4f*)(h + (size_t)n * HID + c) = s;
}

// ---------------- gcn_norm: deg (incl. self loop), dinv = deg^-0.5 ----------------
__global__ void k_deg_init(float* __restrict__ deg) {
    int v = blockIdx.x * blockDim.x + threadIdx.x;
    if (v < NNODES) deg[v] = 1.0f;  // self loop contributes 1
}
__global__ void k_deg_edges(const int* __restrict__ col, float* __restrict__ deg) {
    int e = blockIdx.x * blockDim.x + threadIdx.x;
    if (e < NEDGES) atomicAdd(&deg[col[e]], 1.0f);
}
__global__ void k_dinv(const float* __restrict__ deg, float* __restrict__ dinv) {
    int v = blockIdx.x * blockDim.x + threadIdx.x;
    if (v < NNODES) dinv[v] = rsqrtf(deg[v]);  // deg >= 1 always
}

// ---------------- WMMA f32 GEMM: C[M,N] = A[M,K=100] @ B[K=100,N] (+bias) ----------------
// REQUIRES: M % 16 == 0 (true for 50000 and 2000). N <= NTILES*16.
// B is staged in LDS, zero-padded to NPAD columns, packed as K-pairs (v2f) so each
// lane's B fragment is one ds_load_b64. A fragment is one unconditional global b64 load.
template <int NTILES>
__global__ void __launch_bounds__(256)
k_wmma_gemm(const float* __restrict__ A, const float* __restrict__ B,
            const float* __restrict__ bias, float* __restrict__ C,
            int M, int N, int lda, int ldb, int ldc) {
    constexpr int NPAD = NTILES * 16;
    __shared__ v2f Bs[KDIM / 2][NPAD];  // Bs[j][n] = {B[2j][n], B[2j+1][n]}

    // cooperative, zero-padded staging of all of B (100 x NPAD)
    for (int idx = threadIdx.x; idx < (KDIM / 2) * NPAD; idx += blockDim.x) {
        int j = idx / NPAD;
        int n = idx - j * NPAD;
        v2f bv;
        if (n < N) {
            bv.x = B[(size_t)(2 * j) * ldb + n];
            bv.y = B[(size_t)(2 * j + 1) * ldb + n];
        } else {
            bv.x = 0.0f;
            bv.y = 0.0f;
        }
        Bs[j][n] = bv;
    }
    __syncthreads();

    const int wave = threadIdx.x >> 5;
    const int lane = threadIdx.x & 31;
    const int half = lane >> 4;  // lanes 0-15 vs 16-31
    const int l16  = lane & 15;
    const int m0 = (blockIdx.x * (blockDim.x >> 5) + wave) * 16;
    if (m0 >= M) return;  // wave-uniform -> EXEC stays all-1 for WMMA

    const v8f zero = {0.f, 0.f, 0.f, 0.f, 0.f, 0.f, 0.f, 0.f};
    v8f acc[NTILES];
#pragma unroll
    for (int t = 0; t < NTILES; ++t) acc[t] = zero;

    // A 16x4 fragment (ISA 7.12.2): lane holds row m0+l16, K-pair (k + 2*half, +1).
    // lda=100 floats and K-pair offsets are even -> 8-byte aligned v2f loads.
    const v2f* __restrict__ Arow = (const v2f*)(A + (size_t)(m0 + l16) * lda);

    for (int s = 0; s < KDIM / 4; ++s) {  // 25 steps of K=4
        const int j = 2 * s + half;       // this lane's K-pair index
        v2f a = Arow[j];                  // global_load_b64, unconditional
#pragma unroll
        for (int t = 0; t < NTILES; ++t) {
            v2f b = Bs[j][t * 16 + l16];  // ds_load_b64, unconditional
            acc[t] = __builtin_amdgcn_wmma_f32_16x16x4_f32(
                false, a, false, b, (short)0, acc[t], false, false);
        }
    }

    // C/D 16x16 (ISA 7.12.2): VGPR i holds M = m0 + half*8 + i, N = l16
#pragma unroll
    for (int t = 0; t < NTILES; ++t) {
        const int n = t * 16 + l16;
        if (n >= N) continue;  // only N-edge guard; M is multiple of 16
        const float bv = bias ? bias[n] : 0.0f;
        float* Crow = C + (size_t)(m0 + half * 8) * ldc + n;
#pragma unroll
        for (int i = 0; i < 8; ++i) {
            Crow[(size_t)i * ldc] = acc[t][i] + bv;
        }
    }
}

// ---------------- edge scatter: agg[col,c..c+3] += 1/clip(dinv[r]*dinv[c]*t[row,c..c+3]) ----
// 25 threads per edge, each handles 4 channels: 1 b128 gather + 4 f32 atomics.
// handles E real edges plus N self-loop edges (e >= E)
__global__ void k_edge_scatter(const float* __restrict__ t, const int* __restrict__ row,
                               const int* __restrict__ col, const float* __restrict__ dinv,
                               float* __restrict__ agg) {
    const int total = (NEDGES + NNODES) * (HID / 4);
    int idx = blockIdx.x * blockDim.x + threadIdx.x;
    if (idx >= total) return;
    int e = idx / (HID / 4);
    int c = (idx - e * (HID / 4)) * 4;
    int r, cl;
    if (e < NEDGES) {
        r = row[e];
        cl = col[e];
    } else {
        r = cl = e - NEDGES;  // self loop
    }
    float nrm = dinv[r] * dinv[cl];
    v4f tv = *(const v4f*)(t + (size_t)r * HID + c);  // 16B aligned gather
    float* dst = agg + (size_t)cl * HID + c;
    atomicAdd(dst + 0, clip_inv(nrm * tv.x));  // no-return atomics -> STOREcnt path
    atomicAdd(dst + 1, clip_inv(nrm * tv.y));
    atomicAdd(dst + 2, clip_inv(nrm * tv.z));
    atomicAdd(dst + 3, clip_inv(nrm * tv.w));
}

// ---------------- finalize: hout = 1/clip(agg/cnt) + b (opt. ReLU) ----------------
__global__ void k_finalize(const float* __restrict__ agg, const float* __restrict__ cnt,
                           const float* __restrict__ bias, float* __restrict__ hout,
                           int do_relu) {
    int idx = blockIdx.x * blockDim.x + threadIdx.x;
    if (idx >= NNODES * (HID / 4)) return;
    int n = idx / (HID / 4);
    int c = (idx - n * (HID / 4)) * 4;
    float rc = 1.0f / cnt[n];
    v4f a = *(const v4f*)(agg + (size_t)n * HID + c);
    v4f b = *(const v4f*)(bias + c);
    v4f o;
    o.x = clip_inv(a.x * rc) + b.x;
    o.y = clip_inv(a.y * rc) + b.y;
    o.z = clip_inv(a.z * rc) + b.z;
    o.w = clip_inv(a.w * rc) + b.w;
    if (do_relu) {
        o.x = fmaxf(o.x, 0.0f);
        o.y = fmaxf(o.y, 0.0f);
        o.z = fmaxf(o.z, 0.0f);
        o.w = fmaxf(o.w, 0.0f);
    }
    *(v4f*)(hout + (size_t)n * HID + c) = o;
}

// ---------------- global mean pool ----------------
__global__ void k_pool_sum(const float* __restrict__ h, const int* __restrict__ batch,
                           float* __restrict__ pooled) {
    int idx = blockIdx.x * blockDim.x + threadIdx.x;
    if (idx >= NNODES * (HID / 4)) return;
    int n = idx / (HID / 4);
    int c = (idx - n * (HID / 4)) * 4;
    v4f hv = *(const v4f*)(h + (size_t)n * HID + c);
    float* dst = pooled + (size_t)batch[n] * HID + c;
    atomicAdd(dst + 0, hv.x);
    atomicAdd(dst + 1, hv.y);
    atomicAdd(dst + 2, hv.z);
    atomicAdd(dst + 3, hv.w);
}
__global__ void k_pool_cnt(const int* __restrict__ batch, float* __restrict__ gcnt) {
    int v = blockIdx.x * blockDim.x + threadIdx.x;
    if (v < NNODES) atomicAdd(&gcnt[batch[v]], 1.0f);
}
__global__ void k_pool_div(float* __restrict__ pooled, const float* __restrict__ gcnt) {
    int idx = blockIdx.x * blockDim.x + threadIdx.x;
    if (idx >= NGRAPHS * HID) return;
    pooled[idx] /= fmaxf(gcnt[idx / HID], 1.0f);
}

extern "C" void kernel_launch(void* const* d_in, const int* in_sizes, int n_in,
                              void* d_out, int out_size, void* d_ws, size_t ws_size,
                              hipStream_t stream) {
    const int*   x     = (const int*)d_in[0];
    const int*   ei    = (const int*)d_in[1];   // [2, E]: row = ei, col = ei + E
    const int*   batch = (const int*)d_in[2];
    const float* emb   = (const float*)d_in[3];
    const float* W1    = (const float*)d_in[4];
    const float* b1    = (const float*)d_in[5];
    const float* W2    = (const float*)d_in[6];
    const float* b2    = (const float*)d_in[7];
    const float* Wlin  = (const float*)d_in[8];
    const float* blin  = (const float*)d_in[9];
    float* out = (float*)d_out;

    const int* row = ei;
    const int* col = ei + NEDGES;

    // workspace layout (floats; all regions 16B aligned)
    const size_t NH = (size_t)NNODES * HID;
    float* ws     = (float*)d_ws;
    float* hbuf   = ws;                       // N*H  node features
    float* tbuf   = hbuf + NH;                // N*H  h @ W
    float* aggbuf = tbuf + NH;                // N*H  scatter accumulator
    float* deg    = aggbuf + NH;              // N    (== cnt)
    float* dinv   = deg + NNODES;             // N
    float* pooled = dinv + NNODES;            // G*H
    float* gcnt   = pooled + (size_t)NGRAPHS * HID;  // G   (contiguous after pooled)

    const int BLK = 256;  // 8 wave32 waves per block
    const int NH4    = (int)(NH / 4);                       // 1,250,000
    const int gNH4   = (NH4 + BLK - 1) / BLK;
    const int gEnc   = (NNODES * (HID / 4) + BLK - 1) / BLK;
    const int gNode  = (NNODES + BLK - 1) / BLK;
    const int gEdge  = (NEDGES + BLK - 1) / BLK;
    const int gScat  = ((NEDGES + NNODES) * (HID / 4) + BLK - 1) / BLK;
    const int poolZ4 = (NGRAPHS * HID + NGRAPHS) / 4;       // 50,500 (divisible)
    const int gPoolZ = (poolZ4 + BLK - 1) / BLK;
    const int gPoolD = (NGRAPHS * HID + BLK - 1) / BLK;
    const int gGemmN = ((NNODES / 16) + 7) / 8;   // 8 waves/block, 16 rows/wave
    const int gGemmG = ((NGRAPHS / 16) + 7) / 8;

    // 1) atom encoder
    k_encoder<<<gEnc, BLK, 0, stream>>>(x, emb, hbuf);

    // 2) gcn_norm degrees
    k_deg_init<<<gNode, BLK, 0, stream>>>(deg);
    k_deg_edges<<<gEdge, BLK, 0, stream>>>(col, deg);
    k_dinv<<<gNode, BLK, 0, stream>>>(deg, dinv);

    // 3) layer 1: t = h@W1; scatter; finalize(+ReLU) -> hbuf
    k_wmma_gemm<7><<<gGemmN, BLK, 0, stream>>>(hbuf, W1, nullptr, tbuf,
                                               NNODES, HID, HID, HID, HID);
    k_zero4<<<gNH4, BLK, 0, stream>>>((v4f*)aggbuf, NH4);
    k_edge_scatter<<<gScat, BLK, 0, stream>>>(tbuf, row, col, dinv, aggbuf);
    k_finalize<<<gEnc, BLK, 0, stream>>>(aggbuf, deg, b1, hbuf, 1);

    // 4) layer 2 (no ReLU)
    k_wmma_gemm<7><<<gGemmN, BLK, 0, stream>>>(hbuf, W2, nullptr, tbuf,
                                               NNODES, HID, HID, HID, HID);
    k_zero4<<<gNH4, BLK, 0, stream>>>((v4f*)aggbuf, NH4);
    k_edge_scatter<<<gScat, BLK, 0, stream>>>(tbuf, row, col, dinv, aggbuf);
    k_finalize<<<gEnc, BLK, 0, stream>>>(aggbuf, deg, b2, hbuf, 0);

    // 5) global mean pool
    k_zero4<<<gPoolZ, BLK, 0, stream>>>((v4f*)pooled, poolZ4);  // pooled + gcnt
    k_pool_sum<<<gEnc, BLK, 0, stream>>>(hbuf, batch, pooled);
    k_pool_cnt<<<gNode, BLK, 0, stream>>>(batch, gcnt);
    k_pool_div<<<gPoolD, BLK, 0, stream>>>(pooled, gcnt);

    // 6) out = pooled @ Wlin + blin   [2000 x 128]
    k_wmma_gemm<8><<<gGemmG, BLK, 0, stream>>>(pooled, Wlin, blin, out,
                                               NGRAPHS, NTASKS, HID, NTASKS, NTASKS);
}